// BroadcastConv_10127532884367
// MI455X (gfx1250) — compile-verified
//
#include <hip/hip_runtime.h>

// ---------------------------------------------------------------------------
// Max-plus (tropical) 5x5 convolution, B=4, C=O=16, H=W=64, pad=2 (-inf), f32.
//   out[b,o,y,x] = max_{c,i,j} xpad[b,c,y+i,x+j] + kernel[o,c,4-i,4-j]
// VALU-bound (210 MFLOP vs 2 MB traffic). WMMA cannot express (max,+), so the
// CDNA5 feature used is the Tensor Data Mover: TENSOR_LOAD_TO_LDS descriptors
// DMA each channel tile into LDS (hardware row padding 64->72 f32), double-
// buffered across passes via TENSORcnt (s_wait_tensorcnt). Fallback path:
// GLOBAL_LOAD_ASYNC_TO_LDS_B64 + s_wait_asynccnt.
// ---------------------------------------------------------------------------

#define AS1 __attribute__((address_space(1)))
#define AS3 __attribute__((address_space(3)))

typedef int           v2i __attribute__((ext_vector_type(2)));
typedef int           v4i __attribute__((ext_vector_type(4)));
typedef int           v8i __attribute__((ext_vector_type(8)));
typedef unsigned int  v4u __attribute__((ext_vector_type(4)));

#if defined(__has_builtin)
#if __has_builtin(__builtin_amdgcn_tensor_load_to_lds)
#define HAVE_TDM 1
#endif
#if __has_builtin(__builtin_amdgcn_global_load_async_to_lds_b64)
#define HAVE_ASYNC_LDS 1
#endif
#endif
#ifndef HAVE_TDM
#define HAVE_TDM 0
#endif
#ifndef HAVE_ASYNC_LDS
#define HAVE_ASYNC_LDS 0
#endif

#define NEGINF (-__builtin_inff())

// ---------------------------------------------------------------------------
__global__ void __launch_bounds__(256)
BroadcastConv_maxplus_kernel(const float* __restrict__ imgs,   // (4,16,64,64)
                             const float* __restrict__ kern,   // (16,16,5,5)
                             float* __restrict__ out)          // (4,16,64,64)
{
  constexpr int C = 16, O = 16, H = 64, W = 64;
  constexpr int TILE_H = 16;                 // output rows per block
  constexpr int ROWS   = TILE_H + 4;         // 20 haloed rows
  constexpr int RS     = 72;                 // LDS row stride (floats)
  constexpr int CH_BLK = 4;                  // channels per pass
  constexpr int NP     = C / CH_BLK;         // 4 passes
  constexpr int CH_STRIDE  = ROWS * RS;      // 1440 floats
  constexpr int BUF_FLOATS = CH_BLK * CH_STRIDE;   // 5760 floats = 23 KB
  constexpr int TILE_FLOATS = 2 * BUF_FLOATS;      // double buffered, 46 KB

  __shared__ __align__(16) float s_tile[TILE_FLOATS];
  __shared__ float s_w[C * 25];

  const int tid = threadIdx.x;
  const int o   = blockIdx.x & 15;
  const int rt  = (blockIdx.x >> 4) & 3;
  const int b   = blockIdx.x >> 6;
  const int gr0 = rt * TILE_H;

  // valid haloed-row window (shared by all channels of this block)
  const int lr0   = (gr0 == 0) ? 2 : 0;            // first valid LDS row
  const int g0v   = gr0 - 2 + lr0;                 // first valid image row
  const int glast = (gr0 + TILE_H + 1 < H) ? (gr0 + TILE_H + 1) : (H - 1);
  const int nrows = glast - g0v + 1;               // 18 or 20

  // ---- prefill both buffers with -inf (halo rows keep the pad value) -----
  {
    const float4 negv = make_float4(NEGINF, NEGINF, NEGINF, NEGINF);
    float4* t4 = (float4*)s_tile;
    for (int i = tid; i < TILE_FLOATS / 4; i += 256) t4[i] = negv;
  }
  for (int idx = tid; idx < C * 25; idx += 256) s_w[idx] = kern[o * (C * 25) + idx];
  __syncthreads();

  const int x    = tid & 63;   // output column
  const int rb   = tid >> 6;   // 0..3 : 4 consecutive output rows
  const int wave = tid >> 5;   // wave32 id
  const int lane = tid & 31;

#if HAVE_TDM
  // LDS byte offset of the tile base (AS3 pointers are LDS offsets)
  const unsigned lds_base = (unsigned)(size_t)(AS3 float*)s_tile;

  // Issue one TENSOR_LOAD_TO_LDS per channel of pass `p` into buffer `buf`.
  auto issue_pass = [&](int p, int buf) {
    #pragma unroll
    for (int cl = 0; cl < CH_BLK; ++cl) {
      const int c = p * CH_BLK + cl;
      const unsigned long long ga =
          (unsigned long long)(size_t)(imgs + ((size_t)((b * C + c) * H + g0v)) * W);
      const unsigned la =
          lds_base + (unsigned)((buf * BUF_FLOATS + cl * CH_STRIDE + lr0 * RS + 4) * 4);

      // D# group 0: count=1 | lds_addr | global_addr[56:0] | type=2
      v4u g0;
      g0.x = 1u;                               // count=1 (valid), user mode
      g0.y = la;                               // lds_addr (bytes)
      g0.z = (unsigned)(ga & 0xffffffffu);     // global_addr[31:0]
      g0.w = (unsigned)((ga >> 32) & 0x01ffffffu) | 0x80000000u;  // [56:32]|type=2

      // D# group 1: data_size=4B(2), pad_enable, pad_interval=5 (64 DW),
      //             pad_amount=7 (8 DW) -> LDS row stride 72 floats.
      v8i g1;
      g1[0] = (int)((2u << 16) | (1u << 20) | (5u << 22) | (7u << 25));
      g1[1] = (int)(64u << 16);                // tensor_dim0 = 64 (bits 79:48)
      g1[2] = (int)((unsigned)nrows << 16);    // tensor_dim1 lo16 (bits 95:80)
      g1[3] = (int)(64u << 16);                // tile_dim0 = 64 (bits 127:112)
      g1[4] = nrows;                           // tile_dim1 (bits 143:128), tile_dim2=0
      g1[5] = 64;                              // tensor_dim0_stride lo32 = 64 elements
      g1[6] = 0;                               // stride0 hi16 | stride1 lo16
      g1[7] = 0;                               // stride1 hi32
      v4i gz4 = {0, 0, 0, 0};                  // groups 2/3 unused (2D tile)
      v8i gz8 = {0, 0, 0, 0, 0, 0, 0, 0};      // unused trailing operand
      __builtin_amdgcn_tensor_load_to_lds(g0, g1, gz4, gz4, gz8, /*cpol*/0);
    }
  };
#else
  // Fallback: wave-uniform async row copies (32 lanes x b64 = one 64-f32 row)
  auto issue_pass = [&](int p, int buf) {
    for (int rc = wave; rc < CH_BLK * ROWS; rc += 8) {
      const int cl = rc / ROWS;
      const int lr = rc - cl * ROWS;
      const int g  = gr0 - 2 + lr;
      if (g >= 0 && g < H) {
        const int c = p * CH_BLK + cl;
        const float* gp = imgs + ((size_t)((b * C + c) * H + g)) * W + lane * 2;
        float* lp = s_tile + buf * BUF_FLOATS + cl * CH_STRIDE + lr * RS + 4 + lane * 2;
#if HAVE_ASYNC_LDS
        __builtin_amdgcn_global_load_async_to_lds_b64((AS1 v2i*)gp, (AS3 v2i*)lp, 0, 0);
#else
        const float2 v = *(const float2*)gp;
        *(float2*)lp = v;
#endif
      }
    }
  };
#endif

  float acc[4] = {NEGINF, NEGINF, NEGINF, NEGINF};

#if HAVE_TDM
  if (wave == 0) issue_pass(0, 0);
#else
  issue_pass(0, 0);
#endif

  for (int p = 0; p < NP; ++p) {
    const int buf = p & 1;
#if HAVE_TDM
    if (wave == 0) {
      if (p + 1 < NP) {
        issue_pass(p + 1, buf ^ 1);              // prefetch next pass
#if __has_builtin(__builtin_amdgcn_s_wait_tensorcnt)
        __builtin_amdgcn_s_wait_tensorcnt(CH_BLK);   // current pass done
#else
        asm volatile("s_wait_tensorcnt 4" ::: "memory");
#endif
      } else {
#if __has_builtin(__builtin_amdgcn_s_wait_tensorcnt)
        __builtin_amdgcn_s_wait_tensorcnt(0);
#else
        asm volatile("s_wait_tensorcnt 0" ::: "memory");
#endif
      }
    }
#else
    if (p + 1 < NP) issue_pass(p + 1, buf ^ 1);
#if HAVE_ASYNC_LDS
#if __has_builtin(__builtin_amdgcn_s_wait_asynccnt)
    __builtin_amdgcn_s_wait_asynccnt(0);
#else
    asm volatile("s_wait_asynccnt 0" ::: "memory");
#endif
#endif
#endif
    __syncthreads();   // buffer `buf` for pass p is now fully staged

    // ---- compute: 8-row sliding window over CH_BLK channels --------------
    for (int cl = 0; cl < CH_BLK; ++cl) {
      const int c = p * CH_BLK + cl;
      float wf[25];                      // wf[i*5+j] = kernel[o,c,4-i,4-j]
      #pragma unroll
      for (int i = 0; i < 5; ++i)
        #pragma unroll
        for (int j = 0; j < 5; ++j)
          wf[i * 5 + j] = s_w[c * 25 + (4 - i) * 5 + (4 - j)];

      const float* base =
          s_tile + buf * BUF_FLOATS + cl * CH_STRIDE + (rb * 4) * RS + (x + 2);
      #pragma unroll
      for (int lr = 0; lr < 8; ++lr) {   // xpad rows rb*4 .. rb*4+7
        float v[5];
        #pragma unroll
        for (int j = 0; j < 5; ++j) v[j] = base[lr * RS + j];
        #pragma unroll
        for (int r = 0; r < 4; ++r) {
          const int i = lr - r;
          if (i >= 0 && i < 5) {
            float t0 = v[0] + wf[i * 5 + 0];
            float t1 = v[1] + wf[i * 5 + 1];
            acc[r] = fmaxf(acc[r], fmaxf(t0, t1));   // -> v_max3_num_f32
            float t2 = v[2] + wf[i * 5 + 2];
            float t3 = v[3] + wf[i * 5 + 3];
            acc[r] = fmaxf(acc[r], fmaxf(t2, t3));
            acc[r] = fmaxf(acc[r], v[4] + wf[i * 5 + 4]);
          }
        }
      }
    }
    __syncthreads();   // readers done before this buffer is re-staged
  }

  // ---- write 4 output rows (lanes span consecutive x: coalesced) ---------
  #pragma unroll
  for (int r = 0; r < 4; ++r) {
    const int y = gr0 + rb * 4 + r;
    out[((size_t)((b * O + o) * H + y)) * W + x] = acc[r];
  }
}

extern "C" void kernel_launch(void* const* d_in, const int* in_sizes, int n_in,
                              void* d_out, int out_size, void* d_ws, size_t ws_size,
                              hipStream_t stream) {
  (void)in_sizes; (void)n_in; (void)out_size; (void)d_ws; (void)ws_size;
  const float* imgs = (const float*)d_in[0];   // (4,16,64,64) f32
  const float* kern = (const float*)d_in[1];   // (16,16,5,5)  f32
  // d_in[2] = stride (1), d_in[3] = padding (2): fixed by the reference setup.
  float* out = (float*)d_out;                  // (4,16,64,64) f32

  dim3 grid(256);   // B(4) * row_tiles(4) * O(16)
  dim3 block(256);  // 8 waves (wave32)
  hipLaunchKernelGGL(BroadcastConv_maxplus_kernel, grid, block, 0, stream,
                     imgs, kern, out);
}